// Encoder_13761075216667
// MI455X (gfx1250) — compile-verified
//
#include <hip/hip_runtime.h>
#include <math.h>

typedef __attribute__((ext_vector_type(2))) float v2f;
typedef __attribute__((ext_vector_type(8))) float v8f;

#define B_    4
#define N_    4096
#define C_    60
#define NCAM_ 4
#define NPC   (N_ / NCAM_)        // 1024 tokens per camera
#define M_    (N_ / 5)            // 819 sampled points
#define KNN_  8

// output layout (floats, concatenated in reference return order)
#define OUT_FEATS 0
#define OUT_POS   (B_ * (M_ + NCAM_) * C_)
#define OUT_CAM   (OUT_POS + B_ * (M_ + NCAM_) * 3)

// ---------------------------------------------------------------------------
// Kernel 0: per-row squared norms  |x_i|^2   (B*N rows, C=60)
// ---------------------------------------------------------------------------
__global__ __launch_bounds__(256) void row_norms(const float* __restrict__ feat,
                                                 float* __restrict__ norms) {
    int idx = blockIdx.x * blockDim.x + threadIdx.x;
    if (idx >= B_ * N_) return;
    const float* r = feat + (size_t)idx * C_;
    float s = 0.f;
#pragma unroll
    for (int c = 0; c < C_; ++c) s += r[c] * r[c];
    norms[idx] = s;
}

// ---------------------------------------------------------------------------
// Kernel 1: fused Gram-WMMA + running top-8 smallest distances + density.
// One wave (32 threads) per (batch, 16-row strip). Scans columns in pairs of
// 16x16 tiles: lanes 0-15 maintain top-8 for even tiles, lanes 16-31 for odd
// tiles of the SAME rows; merged via __shfl at the end.
// ---------------------------------------------------------------------------
__global__ __launch_bounds__(32) void knn_density(const float* __restrict__ feat,
                                                  const float* __restrict__ norms,
                                                  float* __restrict__ density) {
    // two 16x16 tiles, row stride 20 floats (16B aligned rows, conflict-light)
    __shared__ float lds[2 * 16 * 20];

    const int b    = blockIdx.y;
    const int i0   = blockIdx.x * 16;
    const int lane = threadIdx.x;   // 0..31
    const int lo   = lane & 15;
    const int hi   = lane >> 4;     // 0 or 1

    const float* fb = feat + (size_t)b * N_ * C_;
    const float* nb = norms + b * N_;

    // ---- A operand (rows i0..i0+15, all 60 features) -> 15 chunks of K=4.
    // ISA layout (f32 16x4 A): lanes 0-15 hold K=4k+0,4k+1; lanes 16-31 K=4k+2,4k+3.
    v2f a[15];
    {
        const float* ar = fb + (size_t)(i0 + lo) * C_ + 2 * hi;
#pragma unroll
        for (int k = 0; k < 15; ++k) a[k] = *(const v2f*)(ar + 4 * k);
    }

    // row norms matching the C/D layout: VGPR v holds rows v (lo half) / v+8 (hi half)
    float rn[8];
#pragma unroll
    for (int v = 0; v < 8; ++v) rn[v] = nb[i0 + v + 8 * hi];

    // per-lane top-8 smallest d^2, ascending (s[7] = current 8th smallest)
    float s[8];
#pragma unroll
    for (int q = 0; q < KNN_; ++q) s[q] = 3.0e38f;

    auto ins = [&](float val) {
        if (val < s[7]) {                 // execz-skipped when no lane qualifies
            float t = val;
#pragma unroll
            for (int i = 6; i >= 0; --i) {
                float up = fmaxf(s[i], t);
                t        = fminf(s[i], t);
                s[i + 1] = up;
            }
            s[0] = t;
        }
    };

    for (int j0 = 0; j0 < N_; j0 += 32) {
        if (j0 + 32 < N_)  // global_prefetch_b8 for next pair of column tiles
            __builtin_prefetch(fb + (size_t)(j0 + 32 + lo) * C_, 0, 1);

        // ---- B operands for column tiles [j0, j0+16) and [j0+16, j0+32)
        const float* br0 = fb + (size_t)(j0 + lo) * C_ + 2 * hi;
        const float* br1 = br0 + (size_t)16 * C_;
        v2f bt0[15], bt1[15];
#pragma unroll
        for (int k = 0; k < 15; ++k) {
            bt0[k] = *(const v2f*)(br0 + 4 * k);
            bt1[k] = *(const v2f*)(br1 + 4 * k);
        }
        float cn0 = nb[j0 + lo];
        float cn1 = nb[j0 + 16 + lo];

        v8f acc0 = {0.f, 0.f, 0.f, 0.f, 0.f, 0.f, 0.f, 0.f};
        v8f acc1 = {0.f, 0.f, 0.f, 0.f, 0.f, 0.f, 0.f, 0.f};
#pragma unroll
        for (int k = 0; k < 15; ++k) {
            acc0 = __builtin_amdgcn_wmma_f32_16x16x4_f32(
                false, a[k], false, bt0[k], (short)0, acc0, false, false);
            acc1 = __builtin_amdgcn_wmma_f32_16x16x4_f32(
                false, a[k], false, bt1[k], (short)0, acc1, false, false);
        }

        // ---- d^2 = |xi|^2 + |xj|^2 - 2*g ; transpose through LDS
#pragma unroll
        for (int v = 0; v < 8; ++v) {
            int row = v + 8 * hi;
            lds[row * 20 + lo]       = rn[v] + cn0 - 2.0f * acc0[v];
            lds[320 + row * 20 + lo] = rn[v] + cn1 - 2.0f * acc1[v];
        }
        __syncthreads();   // single-wave WG: lowers to counter waits

        // lane L owns row (L&15); lo half reads tile0, hi half reads tile1
        const float* myrow = lds + hi * 320 + lo * 20;
#pragma unroll
        for (int q = 0; q < 4; ++q) {
            float4 vv = *(const float4*)(myrow + 4 * q);
            ins(vv.x); ins(vv.y); ins(vv.z); ins(vv.w);
        }
        __syncthreads();
    }

    // ---- merge the two half-lists for each row (lane r <-> lane r+16)
    float pv[KNN_];
#pragma unroll
    for (int q = 0; q < KNN_; ++q) pv[q] = __shfl(s[q], lane ^ 16, 32);
#pragma unroll
    for (int q = 0; q < KNN_; ++q) ins(pv[q]);

    if (hi == 0) {
        float sum = 0.f;
#pragma unroll
        for (int q = 0; q < KNN_; ++q) sum += sqrtf(fmaxf(s[q], 0.f));
        density[b * N_ + i0 + lo] = sum * (1.0f / KNN_);
    }
}

// ---------------------------------------------------------------------------
// Kernel 2: exact top-M by rank (matches jax.lax.top_k ordering incl. ties).
// One block per batch; densities staged in LDS; rank_i = #{j : d_j > d_i or
// (d_j == d_i and j < i)}; element i lands at output slot rank_i if < M.
// ---------------------------------------------------------------------------
__global__ __launch_bounds__(1024) void topk_rank(const float* __restrict__ density,
                                                  int* __restrict__ inds) {
    __shared__ float d[N_];
    const int b = blockIdx.x;
    const float* db = density + b * N_;
    for (int i = threadIdx.x; i < N_; i += 1024) d[i] = db[i];
    __syncthreads();
    for (int i = threadIdx.x; i < N_; i += 1024) {
        float di = d[i];
        int rank = 0;
        for (int j = 0; j < N_; j += 4) {
            float4 v = *(const float4*)(d + j);
            rank += (v.x > di) || (v.x == di && (j + 0) < i);
            rank += (v.y > di) || (v.y == di && (j + 1) < i);
            rank += (v.z > di) || (v.z == di && (j + 2) < i);
            rank += (v.w > di) || (v.w == di && (j + 3) < i);
        }
        if (rank < M_) inds[b * M_ + rank] = i;
    }
}

// ---------------------------------------------------------------------------
// Kernel 3a: per-camera mean tokens (written into the tail of each output row)
// ---------------------------------------------------------------------------
__global__ __launch_bounds__(64) void cam_means(const float* __restrict__ feat,
                                                const float* __restrict__ pcd,
                                                float* __restrict__ out) {
    const int b = blockIdx.x / NCAM_;
    const int cam = blockIdx.x % NCAM_;
    const int t = threadIdx.x;
    if (t < C_) {
        const float* p = feat + ((size_t)b * N_ + (size_t)cam * NPC) * C_ + t;
        float s = 0.f;
        for (int i = 0; i < NPC; ++i) s += p[(size_t)i * C_];
        out[OUT_FEATS + ((size_t)b * (M_ + NCAM_) + M_ + cam) * C_ + t] = s * (1.0f / NPC);
    } else if (t < C_ + 3) {
        const int ch = t - C_;
        const float* p = pcd + ((size_t)b * N_ + (size_t)cam * NPC) * 3 + ch;
        float s = 0.f;
        for (int i = 0; i < NPC; ++i) s += p[i * 3];
        out[OUT_POS + ((size_t)b * (M_ + NCAM_) + M_ + cam) * 3 + ch] = s * (1.0f / NPC);
    }
}

// ---------------------------------------------------------------------------
// Kernel 3b: gather sampled features / positions / camera ids
// ---------------------------------------------------------------------------
__global__ __launch_bounds__(64) void gather_out(const float* __restrict__ feat,
                                                 const float* __restrict__ pcd,
                                                 const int* __restrict__ inds,
                                                 float* __restrict__ out) {
    const int bm = blockIdx.x;
    const int b = bm / M_;
    const int m = bm % M_;
    const int ind = inds[b * M_ + m];
    const int t = threadIdx.x;
    if (t < C_) {
        out[OUT_FEATS + ((size_t)b * (M_ + NCAM_) + m) * C_ + t] =
            feat[((size_t)b * N_ + ind) * C_ + t];
    } else if (t < C_ + 3) {
        out[OUT_POS + ((size_t)b * (M_ + NCAM_) + m) * 3 + (t - C_)] =
            pcd[((size_t)b * N_ + ind) * 3 + (t - C_)];
    } else if (t == 63) {
        out[OUT_CAM + b * M_ + m] = (float)(ind / NPC);
    }
}

// ---------------------------------------------------------------------------
extern "C" void kernel_launch(void* const* d_in, const int* in_sizes, int n_in,
                              void* d_out, int out_size, void* d_ws, size_t ws_size,
                              hipStream_t stream) {
    (void)in_sizes; (void)n_in; (void)out_size; (void)ws_size;
    const float* feat = (const float*)d_in[0];
    const float* pcd  = (const float*)d_in[1];
    float* out = (float*)d_out;

    float* norms   = (float*)d_ws;                 // B*N floats
    float* density = norms + (size_t)B_ * N_;      // B*N floats
    int*   inds    = (int*)(density + (size_t)B_ * N_);  // B*M ints

    row_norms<<<(B_ * N_ + 255) / 256, 256, 0, stream>>>(feat, norms);

    dim3 g1(N_ / 16, B_);
    knn_density<<<g1, 32, 0, stream>>>(feat, norms, density);

    topk_rank<<<B_, 1024, 0, stream>>>(density, inds);

    cam_means<<<B_ * NCAM_, 64, 0, stream>>>(feat, pcd, out);
    gather_out<<<B_ * M_, 64, 0, stream>>>(feat, pcd, inds, out);
}